// TemporalLinearAttention_71262097376064
// MI455X (gfx1250) — compile-verified
//
#include <hip/hip_runtime.h>
#include <math.h>

// Temporal linear attention, fused single kernel for gfx1250 (MI455X).
// out = ((softmax_d(q)*SCALE) @ k_softmaxed_n^T) @ v, with qkv/out projections.
// fp32 WMMA (V_WMMA_F32_16X16X4_F32) end to end: reference is fp32 and the
// kernel is HBM-bound (~268 MB @ 23.3 TB/s), so no precision downcast needed.
// x tile staged to LDS with gfx1250 async-copy (global_load_async_to_lds_b128).

#define NB        4096   // batch*spatial rows
#define NF        16     // frames
#define DIM       512
#define HID       512
#define QKVC      1536   // 3*HID
#define HEADS     8
#define DIM_HEAD  64
#define SCALE_Q   0.125f // DIM_HEAD^-0.5

#define LDX   520        // padded ld (even, mult of 4) for 16x512 buffer
#define LDQ   1544       // padded ld (even) for 16x1536 qkv buffer
#define LDSS  18         // padded ld (even) for per-wave 16x16 S tile

typedef float v2f __attribute__((ext_vector_type(2)));
typedef float v8f __attribute__((ext_vector_type(8)));

__device__ __forceinline__ v8f zero8() {
  v8f z = {0.f, 0.f, 0.f, 0.f, 0.f, 0.f, 0.f, 0.f};
  return z;
}

__device__ __forceinline__ v8f wmma4(v2f a, v2f b, v8f c) {
  // D = A(16x4) * B(4x16) + C(16x16), fp32
  return __builtin_amdgcn_wmma_f32_16x16x4_f32(false, a, false, b, (short)0, c,
                                               false, false);
}

// Async 16B global -> LDS copy (ASYNCcnt-tracked, no VGPR bounce).
// LDS address = low 32 bits of the flat pointer (ISA 10.2 aperture rule).
__device__ __forceinline__ void async_cp_b128(float* lds, const float4* g) {
  const unsigned ldsaddr = (unsigned)(size_t)lds;
  asm volatile("global_load_async_to_lds_b128 %0, %1, off"
               :
               : "v"(ldsaddr), "v"(g)
               : "memory");
}
__device__ __forceinline__ void wait_asynccnt0() {
  asm volatile("s_wait_asynccnt 0x0" ::: "memory");
}

// A fragment (M=16 x K=4), row-major src: A[m][k] = p[m*ld + k].
// Layout (ISA 7.12.2, 32-bit A 16x4): lanes 0-15 hold K=kb+0/kb+1 in v0/v1,
// lanes 16-31 hold K=kb+2/kb+3.
__device__ __forceinline__ v2f fragA(const float* __restrict__ p, int ld,
                                     int kb, int lane) {
  const int m = lane & 15;
  const int sel = (lane < 16) ? 0 : 2;
  const float* q = p + m * ld + kb + sel;
  v2f a;
  a.x = q[0];
  a.y = q[1];
  return a;
}

// B fragment (K=4 x N=16), row-major src: B[k][n] = p[k*ld + n].
__device__ __forceinline__ v2f fragB(const float* __restrict__ p, int ld,
                                     int kb, int n0, int lane) {
  const int n = n0 + (lane & 15);
  const int sel = (lane < 16) ? 0 : 2;
  v2f b;
  b.x = p[(kb + sel) * ld + n];
  b.y = p[(kb + sel + 1) * ld + n];
  return b;
}

// B fragment where B = T^T and T is row-major [N=16 rows x ld]: B[k][n] = p[n*ld + k].
__device__ __forceinline__ v2f fragBT(const float* __restrict__ p, int ld,
                                      int kb, int lane) {
  const int n = lane & 15;
  const int sel = (lane < 16) ? 0 : 2;
  const float* q = p + n * ld + kb + sel;
  v2f b;
  b.x = q[0];
  b.y = q[1];
  return b;
}

// C/D store (16x16), row-major dst: p[m*ld + n0 + n].
// Layout: VGPR r -> row r (lanes 0-15) / row r+8 (lanes 16-31).
__device__ __forceinline__ void storeC(float* __restrict__ p, int ld, int n0,
                                       int lane, v8f c) {
  const int n = n0 + (lane & 15);
  const int mh = (lane < 16) ? 0 : 8;
#pragma unroll
  for (int r = 0; r < 8; ++r) p[(mh + r) * ld + n] = c[r];
}

__global__ __launch_bounds__(256, 1) void tla_fused_kernel(
    const float* __restrict__ x, const float* __restrict__ w_qkv,
    const float* __restrict__ w_out, float* __restrict__ out) {
  extern __shared__ float smem[];
  float* s_x   = smem;                  // 16 x LDX : x tile, later attention out
  float* s_qkv = smem + NF * LDX;       // 16 x LDQ : qkv projection result
  float* s_S   = s_qkv + NF * LDQ;      // 8 waves x 16 x LDSS : S tiles

  const int b    = blockIdx.x;
  const int tid  = threadIdx.x;
  const int wave = tid >> 5;            // 8 waves (wave32)
  const int lane = tid & 31;

  // ---- stage x[b] (16x512 fp32 = 32 KB) into LDS via async copy -----------
  // 2048 float4s, 8 per thread, ASYNCcnt-tracked (no VGPR bounce).
  const float* __restrict__ xb = x + (size_t)b * (NF * DIM);
#pragma unroll
  for (int j = 0; j < 8; ++j) {
    const int i   = tid + j * 256;         // float4 index
    const int row = (i * 4) >> 9;          // /512
    const int col = (i * 4) & (DIM - 1);
    async_cp_b128(s_x + row * LDX + col, ((const float4*)xb) + i);
  }
  wait_asynccnt0();
  __syncthreads();

  // ---- GEMM1: qkv[16x1536] = x_b @ w_qkv (K=512) --------------------------
  // Wave w owns contiguous columns [w*192, (w+1)*192): 3 groups of 4 tiles.
  // One A fragment feeds 4 WMMAs (4x fewer LDS reads); B gathers span 64
  // contiguous columns for good 64B-segment coalescing out of L2.
  for (int g = 0; g < 3; ++g) {
    const int n0 = wave * 192 + g * 64;
    v8f acc[4];
#pragma unroll
    for (int j = 0; j < 4; ++j) acc[j] = zero8();
#pragma unroll 2
    for (int k = 0; k < DIM; k += 4) {
      if ((k & 63) == 0 && (k + 64) < DIM)
        __builtin_prefetch(&w_qkv[(k + 64) * QKVC + n0], 0, 0);
      const v2f a = fragA(s_x, LDX, k, lane);
#pragma unroll
      for (int j = 0; j < 4; ++j) {
        const v2f bb = fragB(w_qkv, QKVC, k, n0 + 16 * j, lane);
        acc[j] = wmma4(a, bb, acc[j]);
      }
    }
#pragma unroll
    for (int j = 0; j < 4; ++j) storeC(s_qkv, LDQ, n0 + 16 * j, lane, acc[j]);
  }
  __syncthreads();

  // ---- per-head attention: wave w <-> head h ------------------------------
  const int h = wave;
  float* qh = s_qkv + h * DIM_HEAD;            // q[m][d] = qh[m*LDQ + d]
  float* kh = s_qkv + HID + h * DIM_HEAD;      // k[n][d] = kh[n*LDQ + d]
  float* vh = s_qkv + 2 * HID + h * DIM_HEAD;  // v[n][e] = vh[n*LDQ + e]

  // softmax over d for each q row (lanes 0..15 own one row each), then *SCALE
  if (lane < NF) {
    float* r = qh + lane * LDQ;
    float mx = r[0];
#pragma unroll
    for (int i = 1; i < DIM_HEAD; ++i) mx = fmaxf(mx, r[i]);
    float s = 0.f;
#pragma unroll
    for (int i = 0; i < DIM_HEAD; ++i) {
      float e = __expf(r[i] - mx);
      r[i] = e;
      s += e;
    }
    const float inv = SCALE_Q / s;
#pragma unroll
    for (int i = 0; i < DIM_HEAD; ++i) r[i] *= inv;
  }
  // softmax over n for each k column (each lane owns 2 of 64 columns)
  for (int c = lane; c < DIM_HEAD; c += 32) {
    float mx = kh[c];
#pragma unroll
    for (int i = 1; i < NF; ++i) mx = fmaxf(mx, kh[i * LDQ + c]);
    float s = 0.f;
#pragma unroll
    for (int i = 0; i < NF; ++i) {
      float e = __expf(kh[i * LDQ + c] - mx);
      kh[i * LDQ + c] = e;
      s += e;
    }
    const float inv = 1.f / s;
#pragma unroll
    for (int i = 0; i < NF; ++i) kh[i * LDQ + c] *= inv;
  }
  __syncthreads();

  // ---- S = q @ k^T  (16x16, K=64) -----------------------------------------
  // (q @ k^T) @ v  ==  q @ (k^T v): associativity, 4x fewer FLOPs, tiny S.
  v8f sacc = zero8();
#pragma unroll
  for (int kk = 0; kk < DIM_HEAD; kk += 4) {
    v2f a  = fragA(qh, LDQ, kk, lane);
    v2f bb = fragBT(kh, LDQ, kk, lane);  // B[k][n] = k[n][k]
    sacc = wmma4(a, bb, sacc);
  }
  float* sS = s_S + wave * NF * LDSS;
  storeC(sS, LDSS, 0, lane, sacc);
  __syncthreads();

  // ---- att_h = S @ v  (16x64, K=16), into recycled s_x buffer -------------
#pragma unroll
  for (int e0 = 0; e0 < DIM_HEAD; e0 += 16) {
    v8f acc = zero8();
#pragma unroll
    for (int kk = 0; kk < NF; kk += 4) {
      v2f a  = fragA(sS, LDSS, kk, lane);
      v2f bb = fragB(vh, LDQ, kk, e0, lane);
      acc = wmma4(a, bb, acc);
    }
    storeC(s_x + h * DIM_HEAD, LDX, e0, lane, acc);  // att[m][h*64+e]
  }
  __syncthreads();

  // ---- GEMM2: out_b[16x512] = att @ w_out (K=512), store to HBM -----------
  // Wave w owns contiguous columns [w*64, (w+1)*64): one group of 4 tiles.
  float* __restrict__ ob = out + (size_t)b * (NF * HID);
  {
    const int n0 = wave * 64;
    v8f acc[4];
#pragma unroll
    for (int j = 0; j < 4; ++j) acc[j] = zero8();
#pragma unroll 2
    for (int k = 0; k < HID; k += 4) {
      if ((k & 63) == 0 && (k + 64) < HID)
        __builtin_prefetch(&w_out[(k + 64) * HID + n0], 0, 0);
      const v2f a = fragA(s_x, LDX, k, lane);
#pragma unroll
      for (int j = 0; j < 4; ++j) {
        const v2f bb = fragB(w_out, HID, k, n0 + 16 * j, lane);
        acc[j] = wmma4(a, bb, acc[j]);
      }
    }
    const int n = n0 + (lane & 15);
    const int mh = (lane < 16) ? 0 : 8;
#pragma unroll
    for (int j = 0; j < 4; ++j)
#pragma unroll
      for (int r = 0; r < 8; ++r) ob[(mh + r) * HID + 16 * j + n] = acc[j][r];
  }
}

extern "C" void kernel_launch(void* const* d_in, const int* in_sizes, int n_in,
                              void* d_out, int out_size, void* d_ws,
                              size_t ws_size, hipStream_t stream) {
  (void)in_sizes; (void)n_in; (void)d_ws; (void)ws_size; (void)out_size;
  const float* x     = (const float*)d_in[0];
  const float* w_qkv = (const float*)d_in[1];
  const float* w_out = (const float*)d_in[2];
  float* out = (float*)d_out;

  const size_t smem_bytes =
      (size_t)(NF * LDX + NF * LDQ + HEADS * NF * LDSS) * sizeof(float);
  tla_fused_kernel<<<NB, 256, smem_bytes, stream>>>(x, w_qkv, w_out, out);
}